// BEVPillarPooling_51659866636799
// MI455X (gfx1250) — compile-verified
//
#include <hip/hip_runtime.h>

typedef __attribute__((ext_vector_type(2))) float v2f;
typedef __attribute__((ext_vector_type(8))) float v8f;

#define HW      (256 * 256)   // pillars
#define ZB      64            // z bins
#define CD      16            // voxel channels (K)
#define DD      64            // BEV dim (N)
#define LN_EPSF 1e-5f

// One wave (32 lanes) processes one pillar:
//   GEMM: M=64 (z), K=16 (C), N=64 (D) via V_WMMA_F32_16X16X4_F32
//   C-matrix initialized with z_part tile -> fuses z_embed@w_e + b1 for free.
__global__ __launch_bounds__(256) void bev_pillar_wmma_kernel(
    const float* __restrict__ dense,    // [HW, 64, 16]
    const float* __restrict__ z_embed,  // [64, 16]
    const float* __restrict__ w1,       // [32, 64]  (rows 0..15 = w_v, 16..31 = w_e)
    const float* __restrict__ b1,       // [64]
    const float* __restrict__ gamma,    // [64]
    const float* __restrict__ beta,     // [64]
    float* __restrict__ out)            // [HW, 64]
{
    __shared__ float zp[ZB * DD];       // 16 KB: z_part = z_embed @ w_e + b1

    const int tid  = threadIdx.x;
    const int lane = tid & 31;
    const int wave = tid >> 5;
    const int half = lane >> 4;         // 0: lanes 0-15, 1: lanes 16-31
    const int l15  = lane & 15;

    // ---- Stage 1: z_part into LDS (tiny, redundant per WG, all L2-hot) ----
    for (int e = tid; e < ZB * DD; e += 256) {
        const int z = e >> 6, d = e & 63;
        float acc = b1[d];
        #pragma unroll
        for (int c = 0; c < CD; ++c)
            acc = fmaf(z_embed[z * CD + c], w1[(CD + c) * DD + d], acc);
        zp[e] = acc;
    }

    // ---- Stage 2: preload B = w_v as 4x4 tiles of 4x16 f32 (2 VGPRs each) ----
    // B layout (4x16): lanes 0-15 hold rows K=0,1; lanes 16-31 hold rows K=2,3.
    v2f bmat[4][4];
    #pragma unroll
    for (int kk = 0; kk < 4; ++kk) {
        #pragma unroll
        for (int nt = 0; nt < 4; ++nt) {
            const int krow = 4 * kk + 2 * half;
            bmat[kk][nt].x = w1[(krow    ) * DD + nt * 16 + l15];
            bmat[kk][nt].y = w1[(krow + 1) * DD + nt * 16 + l15];
        }
    }

    __syncthreads();

    // ---- Stage 3: stream pillar data, WMMA, ReLU, z-accumulate ----
    const int    pillar = blockIdx.x * 8 + wave;
    const float* Abase  = dense + (size_t)pillar * (ZB * CD);

    v8f sum[4];
    #pragma unroll
    for (int nt = 0; nt < 4; ++nt)
        #pragma unroll
        for (int r = 0; r < 8; ++r) sum[nt][r] = 0.f;

    #pragma unroll
    for (int zt = 0; zt < 4; ++zt) {
        // A tile: 16 z-rows x 16 channels. Per k-step: lane loads float2 at
        // row (l15), cols 4*kk + 2*half .. +1  (8B aligned, b64 loads).
        v2f a[4];
        #pragma unroll
        for (int kk = 0; kk < 4; ++kk) {
            const float* p = Abase + (zt * 16 + l15) * CD + 4 * kk + 2 * half;
            a[kk] = *(const v2f*)p;
        }

        #pragma unroll
        for (int nt = 0; nt < 4; ++nt) {
            // C init = z_part tile: element (r, lane) = zp[zt*16 + r + 8*half][nt*16 + l15]
            v8f c;
            #pragma unroll
            for (int r = 0; r < 8; ++r)
                c[r] = zp[(zt * 16 + r + half * 8) * DD + nt * 16 + l15];

            // K=16 as 4 chained 16x16x4 f32 WMMAs
            #pragma unroll
            for (int kk = 0; kk < 4; ++kk)
                c = __builtin_amdgcn_wmma_f32_16x16x4_f32(
                        false, a[kk], false, bmat[kk][nt],
                        (short)0, c, false, false);

            // ReLU then accumulate across z-tiles
            #pragma unroll
            for (int r = 0; r < 8; ++r)
                sum[nt][r] += fmaxf(c[r], 0.f);
        }
    }

    // ---- Stage 4: reduce over rows (z). VGPR r holds M=r (lanes 0-15) / M=r+8
    // (lanes 16-31); same column N in both halves -> in-lane add + xor16. ----
    float s[4];
    #pragma unroll
    for (int nt = 0; nt < 4; ++nt) {
        float t = 0.f;
        #pragma unroll
        for (int r = 0; r < 8; ++r) t += sum[nt][r];
        t += __shfl_xor(t, 16, 32);
        s[nt] = t;      // pillar sum for d = nt*16 + l15 (duplicated across halves)
    }

    // ---- Stage 5: LayerNorm over the 64 dims (16-lane shuffle tree) ----
    float ls = s[0] + s[1] + s[2] + s[3];
    float lq = s[0] * s[0] + s[1] * s[1] + s[2] * s[2] + s[3] * s[3];
    #pragma unroll
    for (int m = 1; m < 16; m <<= 1) {
        ls += __shfl_xor(ls, m, 32);
        lq += __shfl_xor(lq, m, 32);
    }
    const float mean = ls * (1.f / 64.f);
    const float var  = lq * (1.f / 64.f) - mean * mean;
    const float rstd = rsqrtf(var + LN_EPSF);

    if (half == 0) {
        float* obase = out + (size_t)pillar * DD;
        #pragma unroll
        for (int nt = 0; nt < 4; ++nt) {
            const int d = nt * 16 + l15;
            obase[d] = (s[nt] - mean) * rstd * gamma[d] + beta[d];
        }
    }
}

extern "C" void kernel_launch(void* const* d_in, const int* in_sizes, int n_in,
                              void* d_out, int out_size, void* d_ws, size_t ws_size,
                              hipStream_t stream) {
    const float* dense   = (const float*)d_in[0];
    const float* z_embed = (const float*)d_in[1];
    const float* w1      = (const float*)d_in[2];
    const float* b1      = (const float*)d_in[3];
    const float* gamma   = (const float*)d_in[4];
    const float* beta    = (const float*)d_in[5];
    float* out = (float*)d_out;

    const int blocks = HW / 8;   // 8 waves (pillars) per 256-thread block
    bev_pillar_wmma_kernel<<<blocks, 256, 0, stream>>>(
        dense, z_embed, w1, b1, gamma, beta, out);
}